// DSQGAttentionV10_58772332478887
// MI455X (gfx1250) — compile-verified
//
#include <hip/hip_runtime.h>
#include <math.h>

typedef __attribute__((ext_vector_type(2))) float v2f;
typedef __attribute__((ext_vector_type(4))) float v4f;
typedef __attribute__((ext_vector_type(8))) float v8f;

#define DB 2
#define DH 16
#define DN 8192
#define DHD 128
#define DJ 20

// wave32: one wave handles a 16-query tile; 8 waves (256 threads) per block.
__global__ __launch_bounds__(256)
void dsqg_fused_kernel(const float* __restrict__ q,
                       const float* __restrict__ k,
                       const float* __restrict__ v,
                       const float* __restrict__ pb,   // [J,H]
                       const float* __restrict__ se,   // [J,HD]
                       float* __restrict__ out)        // [B,H,N,HD]
{
    constexpr float SC = 0.08838834764831845f; // 1/sqrt(128)
    constexpr int OFFS[DJ] = {1,2,3,4,5,6,7,8,9,11,13,15,16,23,32,64,128,256,512,1024};

    __shared__ float sse[8][16][32];   // per-wave 16 queries x 32 (J padded) = 16 KB

    const int tid  = threadIdx.x;
    const int wave = tid >> 5;
    const int lane = tid & 31;
    const int bh   = blockIdx.y;                 // 0 .. B*H-1
    const int h    = bh & (DH - 1);
    const int n0   = blockIdx.x * 128 + wave * 16;   // tile start (16 queries)
    const size_t base = (size_t)bh * DN * DHD;

    // ---------------- Phase A: qse tile via exact fp32 WMMA ----------------
    // C[m][j] = sum_k Q[n0+m][k] * SE[j][k], J padded 20 -> 32 (2 C tiles)
    {
        const int Mrow = lane & 15;
        const int koff = (lane >> 4) * 2;        // A/B K-pair select per ISA layout
        const int jc   = lane & 15;              // B/C column (N index)
        const int j1   = 16 + jc;
        const int j1c  = (j1 < DJ) ? j1 : (DJ - 1);   // clamp padded rows
        const float* qrow  = q  + base + (size_t)(n0 + Mrow) * DHD;
        const float* serow0 = se + (size_t)jc  * DHD;
        const float* serow1 = se + (size_t)j1c * DHD;

        v8f c0 = {}; v8f c1 = {};
        #pragma unroll
        for (int kb = 0; kb < DHD; kb += 4) {
            v2f a  = *(const v2f*)(qrow   + kb + koff);
            v2f b0 = *(const v2f*)(serow0 + kb + koff);
            v2f b1 = *(const v2f*)(serow1 + kb + koff);
            c0 = __builtin_amdgcn_wmma_f32_16x16x4_f32(false, a, false, b0,
                                                       (short)0, c0, false, false);
            c1 = __builtin_amdgcn_wmma_f32_16x16x4_f32(false, a, false, b1,
                                                       (short)0, c1, false, false);
        }
        // C layout: lane holds rows M = i + (lane>>4)*8, column N = lane&15
        const int hi = lane >> 4;
        #pragma unroll
        for (int i = 0; i < 8; ++i) {
            const int m = i + hi * 8;
            sse[wave][m][jc]      = c0[i];
            sse[wave][m][16 + jc] = c1[i];
        }
    }
    __syncthreads();

    // ------------- Phase B: gathered q.k dots + bias + softmax -------------
    // lane layout: query m = lane>>1, HD half (64 floats) = lane&1
    const int  mq   = lane >> 1;
    const int  half = lane & 1;
    const int  nq   = n0 + mq;
    const float* qh = q + base + (size_t)nq * DHD + half * 64;

    v4f qr[16];
    #pragma unroll
    for (int i = 0; i < 16; ++i) qr[i] = *(const v4f*)(qh + i * 4);

    float s[DJ];
    #pragma unroll
    for (int j = 0; j < DJ; ++j) {
        const int  r     = nq - OFFS[j];
        const bool valid = (r >= 0);
        const int  rc    = valid ? r : 0;
        const float* kh  = k + base + (size_t)rc * DHD + half * 64;
        float d = 0.0f;
        #pragma unroll
        for (int i = 0; i < 16; ++i) {
            v4f kv = *(const v4f*)(kh + i * 4);
            d += qr[i].x * kv.x + qr[i].y * kv.y + qr[i].z * kv.z + qr[i].w * kv.w;
        }
        d += __shfl_xor(d, 1, 32);               // combine the two HD halves
        const float sv = (d + sse[wave][mq][j]) * SC + pb[j * DH + h];
        s[j] = valid ? sv : -__builtin_inff();
    }

    // stable softmax over the 20 offsets (all-invalid row -> zero output)
    float mmax = -__builtin_inff();
    #pragma unroll
    for (int j = 0; j < DJ; ++j) mmax = fmaxf(mmax, s[j]);
    const float msafe = (mmax == -__builtin_inff()) ? 0.0f : mmax;

    float p[DJ];
    float l = 0.0f;
    #pragma unroll
    for (int j = 0; j < DJ; ++j) {
        const float e = (s[j] == -__builtin_inff()) ? 0.0f : __expf(s[j] - msafe);
        p[j] = e;
        l += e;
    }
    const float rl = (l > 0.0f) ? (1.0f / l) : 0.0f;
    #pragma unroll
    for (int j = 0; j < DJ; ++j) p[j] *= rl;

    // ---------------- Phase C: weighted gathered-V accumulation ----------------
    v4f acc[16];
    #pragma unroll
    for (int i = 0; i < 16; ++i) acc[i] = (v4f){0.0f, 0.0f, 0.0f, 0.0f};

    #pragma unroll
    for (int j = 0; j < DJ; ++j) {
        const int  r  = nq - OFFS[j];
        const int  rc = (r >= 0) ? r : 0;
        const float* vh = v + base + (size_t)rc * DHD + half * 64;
        const float pj = p[j];                   // 0 for invalid offsets
        #pragma unroll
        for (int i = 0; i < 16; ++i) {
            v4f vv = *(const v4f*)(vh + i * 4);
            acc[i] += pj * vv;
        }
    }

    float* oh = out + base + (size_t)nq * DHD + half * 64;
    #pragma unroll
    for (int i = 0; i < 16; ++i) *(v4f*)(oh + i * 4) = acc[i];
}

extern "C" void kernel_launch(void* const* d_in, const int* in_sizes, int n_in,
                              void* d_out, int out_size, void* d_ws, size_t ws_size,
                              hipStream_t stream) {
    const float* q  = (const float*)d_in[0];   // [B,H,N,HD]
    const float* k  = (const float*)d_in[1];   // [B,H,N,HD]
    const float* v  = (const float*)d_in[2];   // [B,H,N,HD]
    const float* pb = (const float*)d_in[3];   // [J,H]
    const float* se = (const float*)d_in[4];   // [J,HD]
    float* out = (float*)d_out;                // [B,H,N,HD]

    dim3 grid(DN / 128, DB * DH);              // 64 x 32 blocks
    dim3 block(256);                           // 8 wave32 tiles per block
    dsqg_fused_kernel<<<grid, block, 0, stream>>>(q, k, v, pb, se, out);
}